// _DatasetAwareFinalizer_39135742001288
// MI455X (gfx1250) — compile-verified
//
#include <hip/hip_runtime.h>
#include <math.h>

typedef __attribute__((ext_vector_type(2))) float v2f;
typedef __attribute__((ext_vector_type(8))) float v8f;

#define KH   49      // kernel taps
#define PAD  24      // (KH-1)/2
#define SH   96      // small height
#define SW   128     // small width
#define BH   768     // big height
#define BW   1024    // big width

// Bank-conflict-aware LDS row strides (floats); 64 banks x 4B on CDNA5.
#define BSTR  140    // image buffers: 2*140%64=24 (B half-wave split), 140%64=12 (A row spread)
#define WVSTR 100    // Wv: 100%64=36, gcd(36,64)=4 -> 16 distinct bank groups for b64 A loads
#define WHSTR 136    // WhT: 2*136%64=16 -> half-wave rows hit disjoint bank ranges

// Weight of clamped (edge-padded) convolution: contribution of source index j
// to output index y, for a 1-D axis of length `size`.
__device__ __forceinline__ float band_w(int y, int j, int size,
                                        const float* __restrict__ kern,
                                        const float* __restrict__ pre,
                                        const float* __restrict__ suf) {
    float val = 0.0f;
    if (j == 0) {
        int tt = PAD - y;
        if (tt >= 0) val = pre[tt];
    } else if (j == size - 1) {
        int tt = (size - 1) - y + PAD;
        if (tt <= KH - 1) val = suf[tt];
    } else {
        int t = j - y + PAD;
        if (t >= 0 && t <= KH - 1) val = kern[t];
    }
    return val;
}

extern "C" __global__ void __launch_bounds__(256)
fin_blur_kernel(const float* __restrict__ readout,    // B*96*128
                const float* __restrict__ centerbias, // B*768*1024
                const float* __restrict__ scaling,    // B
                const int*   __restrict__ dsidx,      // B
                const float* __restrict__ dsig,       // nd
                const float* __restrict__ dcbw,       // nd
                const float* __restrict__ dprio,      // nd
                int n_datasets,
                float* __restrict__ small_out,        // B*96*128 (ws path)
                float* __restrict__ big_out,          // B*768*1024 (direct path)
                int write_big)
{
    extern __shared__ float smem[];
    float* buf0 = smem;                       // SH*BSTR : img, then final small result
    float* buf1 = buf0 + SH * BSTR;           // SH*BSTR : vertical-blur intermediate
    float* Wv   = buf1 + SH * BSTR;           // SH*WVSTR: vertical band matrix (96x96)
    float* WhT  = Wv   + SH * WVSTR;          // SW*WHSTR: transposed horiz band (128x128)
    float* kern = WhT  + SW * WHSTR;          // 49
    float* pre  = kern + KH;                  // 49 prefix sums
    float* suf  = pre  + KH;                  // 49 suffix sums
    float* red  = suf  + KH;                  // 256 reduction scratch
    float* scal = red  + 256;                 // [0]=prio, [1]=cbw, [2]=lse

    const int b    = blockIdx.x;
    const int tid  = threadIdx.x;
    const int lane = tid & 31;
    const int wave = tid >> 5;
    const int lo   = lane & 15;   // M (or N) index within tile
    const int hi   = lane >> 4;   // selects K pair / upper C rows

    // ---- Phase 0: per-batch kernel weights, prefix/suffix, prio/cbw ----
    if (tid == 0) {
        int   di    = dsidx[b];
        float sigma = dsig[di] * scaling[b];
        float s = 0.0f;
        for (int k = 0; k < KH; ++k) {
            float g = (float)k - (float)PAD;
            float v = expf(-0.5f * (g / sigma) * (g / sigma));
            kern[k] = v; s += v;
        }
        float inv = 1.0f / s;
        float acc = 0.0f;
        for (int k = 0; k < KH; ++k) { kern[k] *= inv; acc += kern[k]; pre[k] = acc; }
        acc = 0.0f;
        for (int k = KH - 1; k >= 0; --k) { acc += kern[k]; suf[k] = acc; }
        float mean = 0.0f;
        for (int d = 0; d < n_datasets; ++d) mean += dprio[d];
        mean /= (float)n_datasets;
        scal[0] = expf(dprio[di] - mean);
        scal[1] = dcbw[di];
    }

    // ---- Phase 1: stage readout image into LDS (padded rows) ----
    for (int i = tid; i < SH * SW; i += 256)
        buf0[(i / SW) * BSTR + (i % SW)] = readout[(size_t)b * SH * SW + i];
    __syncthreads();

    // ---- Phase 1b: materialize band matrices (one-time, branchy part) ----
    for (int i = tid; i < SH * SH; i += 256) {
        int y = i / SH, j = i % SH;
        Wv[y * WVSTR + j] = band_w(y, j, SH, kern, pre, suf);
    }
    for (int i = tid; i < SW * SW; i += 256) {
        int j = i / SW, x = i % SW;   // WhT[j][x] = Wh[x][j]
        WhT[j * WHSTR + x] = band_w(x, j, SW, kern, pre, suf);
    }
    __syncthreads();

    // ---- Phase 2: vertical blur as WMMA band-matmul: buf1 = Wv * buf0 ----
    // 48 output tiles (6 x 8 of 16x16); 8 waves -> 6 tiles per wave.
    for (int t = 0; t < 6; ++t) {
        int tile = wave * 6 + t;
        int tm = tile >> 3, tn = tile & 7;
        int m0 = tm * 16, n0 = tn * 16;
        int mrow = m0 + lo;
        int ncol = n0 + lo;
        v8f acc = {0.f, 0.f, 0.f, 0.f, 0.f, 0.f, 0.f, 0.f};
        #pragma unroll 4
        for (int k0 = 0; k0 < SH; k0 += 4) {
            int kb = k0 + hi * 2;
            // A (16x4 f32): lane lo holds M=lo; VGPR v holds K = 2*hi + v
            v2f a = *(const v2f*)&Wv[mrow * WVSTR + kb];   // even index -> 8B aligned
            // B (4x16 f32): lane lo holds N=lo; VGPR v holds K = 2*hi + v
            v2f bb;
            bb.x = buf0[kb * BSTR + ncol];
            bb.y = buf0[(kb + 1) * BSTR + ncol];
            acc = __builtin_amdgcn_wmma_f32_16x16x4_f32(
                false, a, false, bb, (short)0, acc, false, false);
        }
        // C/D: lane lo holds N=lo; VGPR r holds M = r + 8*hi
        #pragma unroll
        for (int r = 0; r < 8; ++r)
            buf1[(m0 + r + 8 * hi) * BSTR + ncol] = acc[r];
    }
    __syncthreads();

    // ---- Phase 3: horizontal blur: buf0 = buf1 * Wh^T, fused prio + centerbias ----
    const float prio = scal[0];
    const float cbw  = scal[1];
    for (int t = 0; t < 6; ++t) {
        int tile = wave * 6 + t;
        int tm = tile >> 3, tn = tile & 7;
        int m0 = tm * 16, n0 = tn * 16;
        int mrow = m0 + lo;
        int ncol = n0 + lo;
        v8f acc = {0.f, 0.f, 0.f, 0.f, 0.f, 0.f, 0.f, 0.f};
        #pragma unroll 4
        for (int k0 = 0; k0 < SW; k0 += 4) {
            int kb = k0 + hi * 2;
            v2f a = *(const v2f*)&buf1[mrow * BSTR + kb];  // even index -> 8B aligned
            v2f bb;
            bb.x = WhT[kb * WHSTR + ncol];
            bb.y = WhT[(kb + 1) * WHSTR + ncol];
            acc = __builtin_amdgcn_wmma_f32_16x16x4_f32(
                false, a, false, bb, (short)0, acc, false, false);
        }
        #pragma unroll
        for (int r = 0; r < 8; ++r) {
            int M = m0 + r + 8 * hi;
            // nearest-downsampled centerbias: cb_ds[y][x] = centerbias[b, 8y, 8x]
            float cb = centerbias[(size_t)b * BH * BW + (size_t)(M * 8) * BW + (size_t)(ncol * 8)];
            buf0[M * BSTR + ncol] = acc[r] * prio + cbw * cb;
        }
    }
    __syncthreads();

    // ---- Phase 4: logsumexp over the (replicated) full grid ----
    // LSE_full = log(64) + LSE_small  (8x8 replication)
    float lmax = -INFINITY;
    for (int i = tid; i < SH * SW; i += 256)
        lmax = fmaxf(lmax, buf0[(i / SW) * BSTR + (i % SW)]);
    red[tid] = lmax;
    __syncthreads();
    for (int s = 128; s > 0; s >>= 1) {
        if (tid < s) red[tid] = fmaxf(red[tid], red[tid + s]);
        __syncthreads();
    }
    float mx = red[0];
    __syncthreads();
    float lsum = 0.0f;
    for (int i = tid; i < SH * SW; i += 256)
        lsum += expf(buf0[(i / SW) * BSTR + (i % SW)] - mx);
    red[tid] = lsum;
    __syncthreads();
    for (int s = 128; s > 0; s >>= 1) {
        if (tid < s) red[tid] += red[tid + s];
        __syncthreads();
    }
    if (tid == 0) scal[2] = mx + logf(red[0]) + logf(64.0f);
    __syncthreads();
    const float lse = scal[2];

    // ---- Phase 5: emit ----
    if (!write_big) {
        for (int i = tid; i < SH * SW; i += 256)
            small_out[(size_t)b * SH * SW + i] = buf0[(i / SW) * BSTR + (i % SW)] - lse;
    } else {
        // direct replicated write (fallback if workspace too small)
        float4* outp = (float4*)(big_out + (size_t)b * BH * BW);
        const int total4 = BH * BW / 4;   // 196608
        for (int i = tid; i < total4; i += 256) {
            int Y  = i / (BW / 4);
            int X4 = i % (BW / 4);
            float v = buf0[(Y >> 3) * BSTR + (X4 >> 1)] - lse;  // X = 4*X4, X>>3 == X4>>1
            outp[i] = make_float4(v, v, v, v);
        }
    }
}

extern "C" __global__ void __launch_bounds__(256)
fin_upsample_kernel(const float* __restrict__ small, float* __restrict__ out, int B)
{
    size_t i = (size_t)blockIdx.x * blockDim.x + threadIdx.x;   // one float4 each
    size_t total4 = (size_t)B * BH * BW / 4;
    if (i >= total4) return;
    const int per = BH * BW / 4;                 // 196608
    int b   = (int)(i / per);
    int rem = (int)(i % per);
    int Y   = rem / (BW / 4);
    int X4  = rem % (BW / 4);
    float v = small[(size_t)b * SH * SW + (Y >> 3) * SW + (X4 >> 1)];
    ((float4*)out)[i] = make_float4(v, v, v, v);
}

extern "C" void kernel_launch(void* const* d_in, const int* in_sizes, int n_in,
                              void* d_out, int out_size, void* d_ws, size_t ws_size,
                              hipStream_t stream) {
    const float* readout    = (const float*)d_in[0];
    const float* centerbias = (const float*)d_in[1];
    const float* scaling    = (const float*)d_in[2];
    const int*   dsidx      = (const int*)  d_in[3];
    const float* dsig       = (const float*)d_in[4];
    const float* dcbw       = (const float*)d_in[5];
    const float* dprio      = (const float*)d_in[6];
    const int B  = in_sizes[2];   // scaling_factors length
    const int nd = in_sizes[4];   // dataset_sigmas length
    float* out = (float*)d_out;

    const size_t smemFloats = (size_t)(2 * SH * BSTR + SH * WVSTR + SW * WHSTR
                                       + 3 * KH + 256 + 8);
    const size_t smem = smemFloats * sizeof(float);
    const size_t need = (size_t)B * SH * SW * sizeof(float);

    if (ws_size >= need) {
        fin_blur_kernel<<<B, 256, smem, stream>>>(
            readout, centerbias, scaling, dsidx, dsig, dcbw, dprio, nd,
            (float*)d_ws, nullptr, 0);
        size_t total4 = (size_t)B * BH * BW / 4;
        int threads = 256;
        int blocks  = (int)((total4 + threads - 1) / threads);
        fin_upsample_kernel<<<blocks, threads, 0, stream>>>((const float*)d_ws, out, B);
    } else {
        fin_blur_kernel<<<B, 256, smem, stream>>>(
            readout, centerbias, scaling, dsidx, dsig, dcbw, dprio, nd,
            nullptr, out, 1);
    }
}